// Directiveness_83734682403369
// MI455X (gfx1250) — compile-verified
//
#include <hip/hip_runtime.h>
#include <stdint.h>

// CDNA5 / gfx1250: directiveness loss front-end.
// Stage 1: 1920-pt rDFT as f32 GEMM on V_WMMA_F32_16X16X4_F32.
//  - B operand (frame data) staged via GLOBAL_LOAD_ASYNC_TO_LDS_B128,
//    double-buffered in LDS (ASYNCcnt pipelined), and additionally
//    double-buffered in registers across double-steps so ds waits never
//    block the WMMA pipe.
//  - A operand (basis) packed so each lane streams contiguous b128 loads.

typedef __attribute__((ext_vector_type(2))) float v2f;
typedef __attribute__((ext_vector_type(4))) float v4f;
typedef __attribute__((ext_vector_type(8))) float v8f;

#define WIN_N 1920
#define NFREQ 961
#define FPAD  1024         // 64 freq tiles of 16 rows (rows >= 961 are zero)
#define HALF_N 960         // WIN_N / 2 packed entries per half
#define KC    64           // k-chunk staged per LDS buffer
#define NCHUNK (WIN_N / KC)
#define EPSV  1e-8f
#define TWO_PI 6.28318530717958647692f

// ---------------------------------------------------------------- stage 0
// Packed basis: bas[h][f][i], i = 2m + j  <->  sample n = 4m + 2h + j.
__global__ void build_basis_kernel(float* __restrict__ basC,
                                   float* __restrict__ basS,
                                   const float* __restrict__ window) {
  int idx = blockIdx.x * blockDim.x + threadIdx.x;
  if (idx >= 2 * FPAD * HALF_N) return;
  int h   = idx / (FPAD * HALF_N);
  int rem = idx - h * (FPAD * HALF_N);
  int f   = rem / HALF_N;
  int i   = rem - f * HALF_N;
  int m   = i >> 1;
  int j   = i & 1;
  int n   = 4 * m + 2 * h + j;
  float c = 0.0f, s = 0.0f;
  if (f < NFREQ) {
    int mm = (int)(((long long)f * (long long)n) % WIN_N);
    float th = (float)mm * (TWO_PI / (float)WIN_N);
    float w  = window[n] * (1.0f / (float)WIN_N);
    float sv, cv;
    __sincosf(th, &sv, &cv);
    c =  cv * w;
    s = -sv * w;            // rfft: Im = -sum x sin
  }
  basC[idx] = c;
  basS[idx] = s;
}

// ---------------------------------------------------------------- stage 1
__global__ void __launch_bounds__(256)
dft_direct_kernel(const float* __restrict__ pred,
                  const float* __restrict__ targ,
                  const float* __restrict__ basC,
                  const float* __restrict__ basS,
                  float* __restrict__ outInorm,
                  float* __restrict__ outEnergy,
                  int T, int ntN) {
  __shared__ float lds[2][4 * 16 * KC];    // 2 x 16 KB

  const int tid  = (int)threadIdx.x;
  const int wave = tid >> 5;
  const int lane = tid & 31;

  int bid = (int)blockIdx.x;               // ((ib*ntN)+nt)*8 + fblk
  const int fblk = bid & 7;
  bid >>= 3;
  const int nt  = bid % ntN;
  const int ib  = bid / ntN;               // input*2 + b
  const int inp = ib >> 1;
  const int b   = ib & 1;

  const int lhalf = lane >> 4;
  const int lmod  = lane & 15;
  const int ft    = fblk * 8 + wave;
  const int fA    = ft * 16 + lmod;        // < FPAD

  const float* x = inp ? targ : pred;
  const size_t nsamp  = (size_t)T * WIN_N;
  const size_t chBase = (size_t)b * 4 * nsamp;

  // Packed A row pointers for this lane's half.
  const float* pC = basC + ((size_t)lhalf * FPAD + (size_t)fA) * HALF_N;
  const float* pS = basS + ((size_t)lhalf * FPAD + (size_t)fA) * HALF_N;

  // ---- async-copy setup: 4 rounds x 256 threads x 16B = 16 KB per chunk.
  const float* gsrc[4];
  uint32_t     loff[4];                    // LDS byte offset within a buffer
  const uint32_t ldsBase0 = (uint32_t)(uintptr_t)&lds[0][0];
  const uint32_t bufBytes = (uint32_t)(4 * 16 * KC * sizeof(float));
#pragma unroll
  for (int r = 0; r < 4; ++r) {
    int e  = (r * 256 + tid) * 4;          // float index within chunk
    int c  = e >> 10;                      // channel
    int fr = (e >> 6) & 15;                // frame within tile
    int kk = e & 63;                       // sample within chunk (mult of 4)
    int t  = nt * 16 + fr;
    int tc = t < T ? t : (T - 1);          // clamp loads at the edge tile
    gsrc[r] = x + chBase + (size_t)c * nsamp + (size_t)tc * WIN_N + kk;
    loff[r] = (uint32_t)(((c * 16 + fr) * KC + kk) * sizeof(float));
  }

#define ISSUE_CHUNK(pb, k0)                                                   \
  do {                                                                        \
    _Pragma("unroll")                                                         \
    for (int r = 0; r < 4; ++r) {                                             \
      uint64_t ga = (uint64_t)(uintptr_t)(gsrc[r] + (k0));                    \
      uint32_t la = ldsBase0 + (uint32_t)(pb) * bufBytes + loff[r];           \
      asm volatile("global_load_async_to_lds_b128 %0, %1, off"                \
                   :: "v"(la), "v"(ga) : "memory");                           \
    }                                                                         \
  } while (0)

  v8f aWr = {}, aWi = {};
  v8f a1r = {}, a1i = {};
  v8f a2r = {}, a2i = {};
  v8f a3r = {}, a3i = {};

  ISSUE_CHUNK(0, 0);

  for (int ch = 0; ch < NCHUNK; ++ch) {
    const int k0 = ch * KC;
    if (ch + 1 < NCHUNK) {
      ISSUE_CHUNK((ch + 1) & 1, k0 + KC);
      asm volatile("s_wait_asynccnt 0x4" ::: "memory");   // chunk ch landed
    } else {
      asm volatile("s_wait_asynccnt 0x0" ::: "memory");
    }
    __syncthreads();

    const float* lb  = &lds[ch & 1][0];
    const float* lc0 = lb + (0 * 16 + lmod) * KC + 2 * lhalf;
    const float* lc1 = lb + (1 * 16 + lmod) * KC + 2 * lhalf;
    const float* lc2 = lb + (2 * 16 + lmod) * KC + 2 * lhalf;
    const float* lc3 = lb + (3 * 16 + lmod) * KC + 2 * lhalf;

    // Prime the register pipeline with double-step 0 (backend fuses the
    // +0/+4 pairs into ds_load_2addr_b64).
    v2f b00 = *(const v2f*)(lc0 + 0), b01 = *(const v2f*)(lc0 + 4);
    v2f b10 = *(const v2f*)(lc1 + 0), b11 = *(const v2f*)(lc1 + 4);
    v2f b20 = *(const v2f*)(lc2 + 0), b21 = *(const v2f*)(lc2 + 4);
    v2f b30 = *(const v2f*)(lc3 + 0), b31 = *(const v2f*)(lc3 + 4);

#pragma unroll
    for (int s = 0; s < KC; s += 8) {      // two WMMA K-steps per iteration
      // Prefetch the NEXT double-step's B before issuing current WMMAs,
      // so its ds wait lands a full 16-WMMA block later.
      v2f n00, n01, n10, n11, n20, n21, n30, n31;
      if (s + 8 < KC) {
        n00 = *(const v2f*)(lc0 + s + 8);  n01 = *(const v2f*)(lc0 + s + 12);
        n10 = *(const v2f*)(lc1 + s + 8);  n11 = *(const v2f*)(lc1 + s + 12);
        n20 = *(const v2f*)(lc2 + s + 8);  n21 = *(const v2f*)(lc2 + s + 12);
        n30 = *(const v2f*)(lc3 + s + 8);  n31 = *(const v2f*)(lc3 + s + 12);
      }

      v4f acp = *(const v4f*)(pC + ((k0 + s) >> 1));
      v4f asp = *(const v4f*)(pS + ((k0 + s) >> 1));
      v2f aC0 = {acp.x, acp.y}, aC1 = {acp.z, acp.w};
      v2f aS0 = {asp.x, asp.y}, aS1 = {asp.z, asp.w};

      aWr = __builtin_amdgcn_wmma_f32_16x16x4_f32(false, aC0, false, b00, (short)0, aWr, false, false);
      aWi = __builtin_amdgcn_wmma_f32_16x16x4_f32(false, aS0, false, b00, (short)0, aWi, false, false);
      a1r = __builtin_amdgcn_wmma_f32_16x16x4_f32(false, aC0, false, b10, (short)0, a1r, false, false);
      a1i = __builtin_amdgcn_wmma_f32_16x16x4_f32(false, aS0, false, b10, (short)0, a1i, false, false);
      a2r = __builtin_amdgcn_wmma_f32_16x16x4_f32(false, aC0, false, b20, (short)0, a2r, false, false);
      a2i = __builtin_amdgcn_wmma_f32_16x16x4_f32(false, aS0, false, b20, (short)0, a2i, false, false);
      a3r = __builtin_amdgcn_wmma_f32_16x16x4_f32(false, aC0, false, b30, (short)0, a3r, false, false);
      a3i = __builtin_amdgcn_wmma_f32_16x16x4_f32(false, aS0, false, b30, (short)0, a3i, false, false);

      aWr = __builtin_amdgcn_wmma_f32_16x16x4_f32(false, aC1, false, b01, (short)0, aWr, false, false);
      aWi = __builtin_amdgcn_wmma_f32_16x16x4_f32(false, aS1, false, b01, (short)0, aWi, false, false);
      a1r = __builtin_amdgcn_wmma_f32_16x16x4_f32(false, aC1, false, b11, (short)0, a1r, false, false);
      a1i = __builtin_amdgcn_wmma_f32_16x16x4_f32(false, aS1, false, b11, (short)0, a1i, false, false);
      a2r = __builtin_amdgcn_wmma_f32_16x16x4_f32(false, aC1, false, b21, (short)0, a2r, false, false);
      a2i = __builtin_amdgcn_wmma_f32_16x16x4_f32(false, aS1, false, b21, (short)0, a2i, false, false);
      a3r = __builtin_amdgcn_wmma_f32_16x16x4_f32(false, aC1, false, b31, (short)0, a3r, false, false);
      a3i = __builtin_amdgcn_wmma_f32_16x16x4_f32(false, aS1, false, b31, (short)0, a3i, false, false);

      if (s + 8 < KC) {                    // rotate pipeline (SSA, free)
        b00 = n00; b01 = n01; b10 = n10; b11 = n11;
        b20 = n20; b21 = n21; b30 = n30; b31 = n31;
      }
    }
    __syncthreads();                       // buffer reuse fence
  }
#undef ISSUE_CHUNK

  // C/D layout: VGPR r -> row M = r + 8*lhalf, col N = lmod.
  const int    fOutBase = ft * 16 + 8 * lhalf;
  const int    tOut     = nt * 16 + lmod;
  const bool   tok      = (tOut < T);
  const size_t rowBase  = (size_t)ib * NFREQ * (size_t)T;

#pragma unroll
  for (int r = 0; r < 8; ++r) {
    float wr = aWr[r], wi = aWi[r];
    float x1r = a1r[r], x1i = a1i[r];
    float x2r = a2r[r], x2i = a2i[r];
    float x3r = a3r[r], x3i = a3i[r];
    const float S = 1.22474487139158905f;  // sqrt(1.5)
    float i1 = (wr * x1r + wi * x1i) * S;
    float i2 = (wr * x2r + wi * x2i) * S;
    float i3 = (wr * x3r + wi * x3i) * S;
    float inorm  = __builtin_sqrtf(i1 * i1 + i2 * i2 + i3 * i3 + EPSV);
    float energy = 0.5f * (wr * wr + wi * wi +
                           1.5f * (x1r * x1r + x1i * x1i +
                                   x2r * x2r + x2i * x2i +
                                   x3r * x3r + x3i * x3i));
    int fOut = fOutBase + r;
    if (tok && fOut < NFREQ) {
      size_t idx = rowBase + (size_t)fOut * T + tOut;
      outInorm[idx]  = inorm;
      outEnergy[idx] = energy;
    }
  }
}

// ---------------------------------------------------------------- stage 2
__global__ void iir_kernel(const float* __restrict__ inorm,
                           const float* __restrict__ energy,
                           const float* __restrict__ alpha,
                           float* __restrict__ dval, int T) {
  int id = blockIdx.x * blockDim.x + threadIdx.x;
  if (id >= 4 * NFREQ) return;
  int ib  = id / NFREQ;                    // input*2 + b
  int f   = id - ib * NFREQ;
  int inp = ib >> 1;
  int b   = ib & 1;

  float a  = alpha[f];
  float om = 1.0f - a;
  const float* si = inorm  + (size_t)id * T;
  const float* se = energy + (size_t)id * T;
  float* dv = dval + ((size_t)((b * 2 + inp) * NFREQ) + f) * (size_t)T;

  float s0 = 0.0f, s1 = 0.0f;
  dv[0] = 0.001f;                          // t=0: y=0 -> clipped
  for (int t = 1; t < T; ++t) {
    s0 = a * s0 + om * si[t];
    s1 = a * s1 + om * se[t];
    float d = s0 / (s1 + EPSV);
    d = fminf(fmaxf(d, 0.001f), 0.999f);
    dv[t] = d;
  }
}

// ---------------------------------------------------------------- stage 3
__global__ void erb_kernel(const float* __restrict__ m,
                           const float* __restrict__ dval,
                           float* __restrict__ out, int K, int T) {
  int bk = (int)blockIdx.x;                // (b*2 + d)*K + k
  int k  = bk % K;
  int bd = bk / K;
  const float* mk = m + (size_t)k * NFREQ;
  for (int t = (int)threadIdx.x; t < T; t += (int)blockDim.x) {
    const float* dv = dval + (size_t)bd * NFREQ * (size_t)T + t;
    float acc = 0.0f;
    for (int f = 0; f < NFREQ; ++f)
      acc += mk[f] * dv[(size_t)f * T];
    out[(size_t)bk * T + t] = acc;
  }
}

// ---------------------------------------------------------------- launch
extern "C" void kernel_launch(void* const* d_in, const int* in_sizes, int n_in,
                              void* d_out, int out_size, void* d_ws, size_t ws_size,
                              hipStream_t stream) {
  (void)n_in; (void)out_size; (void)ws_size;
  const float* pred   = (const float*)d_in[0];
  const float* targ   = (const float*)d_in[1];
  const float* window = (const float*)d_in[2];
  const float* alpha  = (const float*)d_in[3];
  const float* erbm   = (const float*)d_in[4];
  float* out = (float*)d_out;

  const int nsampC = in_sizes[0] / 8;               // samples per (b, channel)
  const int T      = (nsampC - WIN_N) / WIN_N + 1;  // 1000
  const int K      = in_sizes[4] / NFREQ;           // ERB band count

  // workspace (floats): packed basis cos | sin | inorm | energy | dval
  float* ws = (float*)d_ws;
  const size_t BAS = (size_t)2 * FPAD * HALF_N;
  float* basC   = ws;
  float* basS   = basC + BAS;
  float* inorm  = basS + BAS;
  float* energy = inorm  + (size_t)4 * NFREQ * T;
  float* dval   = energy + (size_t)4 * NFREQ * T;

  hipLaunchKernelGGL(build_basis_kernel,
                     dim3(((unsigned)(2 * FPAD * HALF_N) + 255) / 256), dim3(256), 0, stream,
                     basC, basS, window);

  const int ntN = (T + 15) / 16;
  const int blocks = 4 * ntN * 8;                   // (ib, nt, fblk)
  hipLaunchKernelGGL(dft_direct_kernel,
                     dim3(blocks), dim3(256), 0, stream,
                     pred, targ, basC, basS, inorm, energy, T, ntN);

  hipLaunchKernelGGL(iir_kernel,
                     dim3((4 * NFREQ + 127) / 128), dim3(128), 0, stream,
                     inorm, energy, alpha, dval, T);

  hipLaunchKernelGGL(erb_kernel,
                     dim3(4 * K), dim3(256), 0, stream,
                     erbm, dval, out, K, T);
}